// GCN_52931176956522
// MI455X (gfx1250) — compile-verified
//
#include <hip/hip_runtime.h>
#include <stdint.h>

#define DIM 128

typedef __attribute__((ext_vector_type(2))) float v2f;
typedef __attribute__((ext_vector_type(8))) float v8f;
typedef uint32_t u32x4 __attribute__((ext_vector_type(4)));
typedef uint32_t u32x8 __attribute__((ext_vector_type(8)));

// ---------------------------------------------------------------------------
// Degree / normalization
// ---------------------------------------------------------------------------
__global__ __launch_bounds__(256) void gcn_zero_deg(unsigned* __restrict__ deg, int n) {
  int i = blockIdx.x * 256 + threadIdx.x;
  if (i < n) deg[i] = 0u;
}

__global__ __launch_bounds__(256) void gcn_count_deg(const int* __restrict__ dst,
                                                     unsigned* __restrict__ deg, int e) {
  int i = blockIdx.x * 256 + threadIdx.x;
  if (i < e) atomicAdd(&deg[dst[i]], 1u);
}

// deg buffer reinterpreted in place: u32 count -> f32 rsqrt(deg + 1)
__global__ __launch_bounds__(256) void gcn_dis(unsigned* __restrict__ degdis, int n) {
  int i = blockIdx.x * 256 + threadIdx.x;
  if (i < n) {
    float d = (float)(degdis[i] + 1u);          // +1 self loop, always >= 1
    ((float*)degdis)[i] = rsqrtf(d);
  }
}

// ---------------------------------------------------------------------------
// Dense transform: H[nPad,128] = A[n,128] @ W[128,128] via V_WMMA_F32_16X16X4_F32
// 256 threads = 8 waves, each wave computes a 16-row x 128-col strip.
// W is staged into LDS by the Tensor Data Mover (one 2D D# descriptor).
// Out must have nPad = gridDim.x*128 rows (workspace is padded); loads clamp.
// ---------------------------------------------------------------------------
__global__ __launch_bounds__(256) void gcn_gemm_wmma(const float* __restrict__ A,
                                                     const float* __restrict__ W,
                                                     float* __restrict__ Out,
                                                     int nrows) {
  __shared__ float sW[DIM * DIM];               // 64 KB
  const int tid = threadIdx.x;

  // --- TDM: async-load the 128x128 f32 W tile into LDS (wave 0 issues) ----
  if (tid < 32) {                               // wave-uniform branch
    uint64_t ga  = (uint64_t)(uintptr_t)W;      // global byte address
    uint32_t lds = (uint32_t)(uintptr_t)&sW[0]; // low 32 bits = LDS offset

    u32x4 g0;                                   // D# group 0 (128b)
    g0[0] = 1u;                                 // count=1, no gather/restore
    g0[1] = lds;                                // lds_addr
    g0[2] = (uint32_t)ga;                       // global_addr[31:0]
    g0[3] = (uint32_t)((ga >> 32) & 0x1FFFFFFu) // global_addr[56:32]
            | (2u << 30);                       // type=2 ("image")

    u32x8 g1;                                   // D# group 1 (256b)
    g1[0] = 2u << 16;                           // data_size=2 (4 bytes)
    g1[1] = 128u << 16;                         // tensor_dim0 = 128
    g1[2] = 128u << 16;                         // tensor_dim1 = 128
    g1[3] = 128u << 16;                         // tile_dim0 = 128
    g1[4] = 128u;                               // tile_dim1 = 128 (tile_dim2=0)
    g1[5] = 128u;                               // tensor_dim0_stride = 128
    g1[6] = 0u;
    g1[7] = 0u;

    asm volatile("tensor_load_to_lds %0, %1" :: "s"(g0), "s"(g1) : "memory");
    __builtin_amdgcn_s_wait_tensorcnt(0);
  }
  __syncthreads();

  const int wave  = tid >> 5;                   // 0..7
  const int lane  = tid & 31;
  const int mlane = lane & 15;                  // row / col index within tile
  const int khalf = lane >> 4;                  // 0: K=k..k+1, 1: K=k+2..k+3

  const int rowBase = blockIdx.x * 128 + wave * 16;
  int row  = rowBase + mlane;
  int rowC = row < nrows ? row : (nrows - 1);   // clamp loads (x is not padded)

  v8f acc[8];
#pragma unroll
  for (int ct = 0; ct < 8; ++ct) acc[ct] = (v8f){0.f,0.f,0.f,0.f,0.f,0.f,0.f,0.f};

  const float* ap = A + (size_t)rowC * DIM;
  __builtin_prefetch(ap, 0, 0);                 // global_prefetch_b8 for the row

  for (int k = 0; k < DIM; k += 4) {
    // A fragment (16x4 f32): lanes 0-15 -> K=k,k+1 ; lanes 16-31 -> K=k+2,k+3
    v2f afrag;
    afrag.x = ap[k + khalf * 2 + 0];
    afrag.y = ap[k + khalf * 2 + 1];

#pragma unroll
    for (int ct = 0; ct < 8; ++ct) {
      // B fragment (4x16 f32): lane = column, VGPRs span K (mirrors A layout)
      const int col = ct * 16 + mlane;
      const int bk  = k + khalf * 2;
      v2f bfrag;
      bfrag.x = sW[(bk + 0) * DIM + col];
      bfrag.y = sW[(bk + 1) * DIM + col];

      acc[ct] = __builtin_amdgcn_wmma_f32_16x16x4_f32(
          /*neg_a=*/false, afrag, /*neg_b=*/false, bfrag,
          /*c_mod=*/(short)0, acc[ct], /*reuse_a=*/false, /*reuse_b=*/false);
    }
  }

  // D layout: VGPR j -> (lanes 0-15: M=j, lanes 16-31: M=j+8), N = lane&15.
  // Out is padded to gridDim.x*128 rows -> unconditional stores, no exec dance.
#pragma unroll
  for (int ct = 0; ct < 8; ++ct) {
#pragma unroll
    for (int j = 0; j < 8; ++j) {
      int r = rowBase + j + khalf * 8;
      Out[(size_t)r * DIM + ct * 16 + mlane] = acc[ct][j];
    }
  }
}

// ---------------------------------------------------------------------------
// Aggregation: self-loop + bias init (fully overwrites destination), then
// edge scatter with hardware f32 atomics.
// ---------------------------------------------------------------------------
__global__ __launch_bounds__(256) void gcn_agg_init(const float* __restrict__ h,
                                                    const float* __restrict__ dis,
                                                    const float* __restrict__ bias,
                                                    float* __restrict__ out, int n) {
  int t    = blockIdx.x * 256 + threadIdx.x;    // n * 32 threads, float4 each
  int node = t >> 5;
  int lane = t & 31;
  if (node >= n) return;
  float s = dis[node];
  float w = s * s;                              // self-loop norm = dis^2
  float4 v  = ((const float4*)(h + (size_t)node * DIM))[lane];
  float4 bb = ((const float4*)bias)[lane];
  float4 r;
  r.x = w * v.x + bb.x;
  r.y = w * v.y + bb.y;
  r.z = w * v.z + bb.z;
  r.w = w * v.w + bb.w;
  ((float4*)(out + (size_t)node * DIM))[lane] = r;
}

__global__ __launch_bounds__(256) void gcn_agg_edges(const int* __restrict__ src,
                                                     const int* __restrict__ dst,
                                                     const float* __restrict__ dis,
                                                     const float* __restrict__ h,
                                                     float* __restrict__ out, int e) {
  int t    = blockIdx.x * 256 + threadIdx.x;    // one wave (32 lanes) per edge
  int edge = t >> 5;
  int lane = t & 31;
  if (edge >= e) return;
  // edge is wave-uniform: force scalarization of indices and norm weight
  int s = __builtin_amdgcn_readfirstlane(src[edge]);
  int d = __builtin_amdgcn_readfirstlane(dst[edge]);
  float w = dis[s] * dis[d];
  w = __builtin_bit_cast(float, __builtin_amdgcn_readfirstlane(__builtin_bit_cast(int, w)));
  float4 v = ((const float4*)(h + (size_t)s * DIM))[lane];
  float* o = out + (size_t)d * DIM + lane * 4;
  unsafeAtomicAdd(o + 0, w * v.x);
  unsafeAtomicAdd(o + 1, w * v.y);
  unsafeAtomicAdd(o + 2, w * v.z);
  unsafeAtomicAdd(o + 3, w * v.w);
}

// ---------------------------------------------------------------------------
// Launch
// ---------------------------------------------------------------------------
extern "C" void kernel_launch(void* const* d_in, const int* in_sizes, int n_in,
                              void* d_out, int out_size, void* d_ws, size_t ws_size,
                              hipStream_t stream) {
  const float* x    = (const float*)d_in[0];
  const int*   eidx = (const int*)d_in[1];
  const float* W    = (const float*)d_in[2];
  const float* bias = (const float*)d_in[3];
  float*       out  = (float*)d_out;

  const int n = in_sizes[0] / DIM;              // 50000
  const int e = in_sizes[1] / 2;                // 640000
  const int* src = eidx;                        // edge_index[0]
  const int* dst = eidx + e;                    // edge_index[1]

  const int nPad = (n + 127) & ~127;            // GEMM output padded rows

  // workspace layout (all f32/u32 words)
  unsigned* deg = (unsigned*)d_ws;              // n words: u32 count -> f32 dis
  float* dis = (float*)deg;
  float* h   = (float*)d_ws + ((n + 255) & ~255);
  float* tmp = h + (size_t)nPad * DIM;

  const int gN   = (n + 255) / 256;
  const int gE   = (e + 255) / 256;
  const int gMM  = nPad / 128;                  // 128 rows per block
  const int gNI  = (n * 32 + 255) / 256;        // agg_init: 32 threads/node
  const int gEA  = (e * 32 + 255) / 256;        // agg_edges: 32 threads/edge

  // normalization
  gcn_zero_deg <<<gN, 256, 0, stream>>>(deg, n);
  gcn_count_deg<<<gE, 256, 0, stream>>>(dst, deg, e);
  gcn_dis      <<<gN, 256, 0, stream>>>(deg, n);

  // layer 1: h = x @ W ; tmp = A_norm * h + b
  gcn_gemm_wmma<<<gMM, 256, 0, stream>>>(x, W, h, n);
  gcn_agg_init <<<gNI, 256, 0, stream>>>(h, dis, bias, tmp, n);
  gcn_agg_edges<<<gEA, 256, 0, stream>>>(src, dst, dis, h, tmp, e);

  // layer 2: h = tmp @ W ; out = A_norm * h + b
  gcn_gemm_wmma<<<gMM, 256, 0, stream>>>(tmp, W, h, n);
  gcn_agg_init <<<gNI, 256, 0, stream>>>(h, dis, bias, out, n);
  gcn_agg_edges<<<gEA, 256, 0, stream>>>(src, dst, dis, h, out, e);
}